// CopNet_52003464020600
// MI455X (gfx1250) — compile-verified
//
#include <hip/hip_runtime.h>
#include <math.h>

// Problem constants (from reference setup_inputs):
//   n = 32 mixture comps, d = 16 dims, M = 131072 queries, Ns = 1000 samples
// Knot tables: gk/sk length 1002, mg/dsdg length 1001; padded to 1024 each.
// Workspace layout (floats):
//   [0 .. 16*4096)      per-dim tables: gk | sk | mg | dsdg (1024 each)
//   [16*4096 .. +8192)  per-wave partial sums of log(dens)

#define N_MIX   32
#define N_DIM   16
#define N_QRY   131072
#define N_SMP   1000
#define TAB_STRIDE 4096
#define N_PART  8192   // 1024 blocks * 8 waves

typedef float v2f __attribute__((ext_vector_type(2)));
typedef float v8f __attribute__((ext_vector_type(8)));

__device__ __forceinline__ float fast_rcp(float x) {
  return __builtin_amdgcn_rcpf(x);   // v_rcp_f32
}

// ---------------------------------------------------------------------------
// Kernel 1: per-dim bitonic sort + spline knot table construction.
// One block (1024 threads) per dimension. Sorting data[:,k] sorts g and
// sigmoid(data) simultaneously (both strictly monotone in x).
// ---------------------------------------------------------------------------
__global__ __launch_bounds__(1024) void build_tables(
    const float* __restrict__ data, const float* __restrict__ W,
    const float* __restrict__ b, const float* __restrict__ a,
    float* __restrict__ ws)
{
  const int k = blockIdx.x;   // dimension
  const int t = threadIdx.x;  // 0..1023

  __shared__ float xs[1024];
  __shared__ float gks[1002];
  __shared__ float sks[1002];
  __shared__ float eWk[N_MIX], bk[N_MIX], eas[N_MIX];
  __shared__ float sum_ea_s;

  xs[t] = (t < N_SMP) ? data[t * N_DIM + k] : 3.0e38f;  // +inf pad
  if (t < N_MIX) {
    eWk[t] = __expf(W[t * N_DIM + k]);
    bk[t]  = b[t * N_DIM + k];
    eas[t] = __expf(a[t]);
  }
  __syncthreads();
  if (t == 0) {
    float s = 0.f;
    for (int i = 0; i < N_MIX; ++i) s += eas[i];
    sum_ea_s = s;
  }

  // Bitonic sort of 1024 elements in LDS.
  for (int ksz = 2; ksz <= 1024; ksz <<= 1) {
    for (int j = ksz >> 1; j > 0; j >>= 1) {
      int ixj = t ^ j;
      if (ixj > t) {
        float A = xs[t], B = xs[ixj];
        bool asc = ((t & ksz) == 0);
        if (asc ? (A > B) : (A < B)) { xs[t] = B; xs[ixj] = A; }
      }
      __syncthreads();
    }
  }

  // Knot arrays: endpoints {0,1} plus sorted interior values.
  if (t == 0) { gks[0] = 0.f; sks[0] = 0.f; gks[1001] = 1.f; sks[1001] = 1.f; }
  if (t < N_SMP) {
    float x = xs[t];
    float gsum = 0.f;
#pragma unroll
    for (int i = 0; i < N_MIX; ++i) {
      float Ag = eWk[i] * x + bk[i];
      float sg = fast_rcp(1.f + __expf(-Ag));
      gsum += eas[i] * sg;
    }
    gks[t + 1] = gsum * fast_rcp(sum_ea_s);
    sks[t + 1] = fast_rcp(1.f + __expf(-x));
  }
  __syncthreads();

  float* base = ws + (size_t)k * TAB_STRIDE;
  if (t < 1002) {
    base[t]        = gks[t];
    base[1024 + t] = sks[t];
  }
  if (t < 1001) {
    float g0 = gks[t], g1 = gks[t + 1];
    float s0 = sks[t], s1 = sks[t + 1];
    base[2048 + t] = 0.5f * (g0 + g1);               // mg (midpoints)
    base[3072 + t] = (s1 - s0) * fast_rcp(g1 - g0);  // dsdg (slopes)
  }
}

// ---------------------------------------------------------------------------
// Kernel 2: main per-query kernel. 256 threads = 8 waves; each wave owns a
// 16-query tile (lane halves duplicate phase-1 so phase-2 lane layout matches
// the V_WMMA_F32_16X16X4_F32 A-operand layout exactly: lanes 0-15 hold rows
// M with K=c*4+{0,1}, lanes 16-31 the same rows with K=c*4+{2,3}).
// ---------------------------------------------------------------------------
__global__ __launch_bounds__(256) void copnet_main(
    const float* __restrict__ u, const float* __restrict__ W,
    const float* __restrict__ b, const float* __restrict__ a,
    const float* __restrict__ tabs, float* __restrict__ partials)
{
  __shared__ float gk_s[1024], sk_s[1024], mg_s[1024], dg_s[1024];
  __shared__ float eW_s[N_MIX * N_DIM], b_s[N_MIX * N_DIM], ea_s[N_MIX];

  const int t    = threadIdx.x;
  const int lane = t & 31;
  const int wave = t >> 5;
  const int hi   = (lane >> 4) & 1;   // lane half -> K sub-position
  const int row  = lane & 15;         // WMMA row (query within tile)

  for (int idx = t; idx < N_MIX * N_DIM; idx += 256) {
    eW_s[idx] = __expf(W[idx]);
    b_s[idx]  = b[idx];
  }
  if (t < N_MIX) ea_s[t] = __expf(a[t]);

  const int m = blockIdx.x * 128 + wave * 16 + row;  // query id

  float uq[N_DIM];
#pragma unroll
  for (int k = 0; k < N_DIM; ++k) uq[k] = u[m * N_DIM + k];

  float zu[N_DIM], zdu[N_DIM];

  // ---- Phase 1: stage each dim's tables in LDS, binary-search interp ----
  for (int k = 0; k < N_DIM; ++k) {
    const float* base = tabs + (size_t)k * TAB_STRIDE;
    if (k + 1 < N_DIM)
      __builtin_prefetch(tabs + (size_t)(k + 1) * TAB_STRIDE, 0, 1);
    __syncthreads();
    for (int idx = t; idx < 1024; idx += 256) {
      gk_s[idx] = base[idx];
      sk_s[idx] = base[1024 + idx];
      mg_s[idx] = base[2048 + idx];
      dg_s[idx] = base[3072 + idx];
    }
    __syncthreads();

    float x = uq[k];

    // interp u on (gk, sk), K = 1002, searchsorted(right)-1
    float xq = fminf(fmaxf(x, gk_s[0]), gk_s[1001]);
    int lo = 0, hc = 1002;
    while (lo < hc) { int mid = (lo + hc) >> 1; if (gk_s[mid] <= xq) lo = mid + 1; else hc = mid; }
    int i0 = lo - 1; if (i0 < 0) i0 = 0; if (i0 > 1000) i0 = 1000;
    float g0 = gk_s[i0], g1 = gk_s[i0 + 1];
    float s0 = sk_s[i0], s1 = sk_s[i0 + 1];
    float wi = (xq - g0) * fast_rcp(g1 - g0);
    float st = s0 + wi * (s1 - s0);
    zu[k] = __logf(st) - __logf(1.f - st);   // invsigmoid

    // interp u on (mg, dsdg), K = 1001
    float xq2 = fminf(fmaxf(x, mg_s[0]), mg_s[1000]);
    lo = 0; hc = 1001;
    while (lo < hc) { int mid = (lo + hc) >> 1; if (mg_s[mid] <= xq2) lo = mid + 1; else hc = mid; }
    int i1 = lo - 1; if (i1 < 0) i1 = 0; if (i1 > 999) i1 = 999;
    float mA = mg_s[i1], mB = mg_s[i1 + 1];
    float dA = dg_s[i1], dB = dg_s[i1 + 1];
    float w2 = (xq2 - mA) * fast_rcp(mB - mA);
    float sd = dA + w2 * (dB - dA);
    zdu[k] = sd * fast_rcp(st * (1.f - st));
  }
  __syncthreads();

  // ---- Phase 2: AAA products + WMMA reduction over 32 mixture comps ----
  // dens[m] = sum_i ea[i] * prod_k sig'(A)*eW*zdu   via 8x wmma 16x16x4.
  v8f cacc = {0.f, 0.f, 0.f, 0.f, 0.f, 0.f, 0.f, 0.f};
  for (int c = 0; c < 8; ++c) {
    float av0, av1;
#pragma unroll
    for (int ii = 0; ii < 2; ++ii) {
      const int i = c * 4 + 2 * hi + ii;
      float acc = 1.f;
#pragma unroll
      for (int k = 0; k < N_DIM; ++k) {
        float ew = eW_s[i * N_DIM + k];
        float A  = ew * zu[k] + b_s[i * N_DIM + k];
        float e  = __expf(-A);                 // native v_exp_f32
        float r  = fast_rcp(1.f + e);          // native v_rcp_f32
        float dp = e * r * r;                  // sigmoid'(A) = sA*(1-sA)
        acc *= dp * ew * zdu[k];
      }
      if (ii == 0) av0 = acc; else av1 = acc;
    }
    v2f a2 = {av0, av1};
    // B[K, n] = ea[chunk K] replicated across columns (same K layout as A).
    v2f b2 = {ea_s[c * 4 + 2 * hi + 0], ea_s[c * 4 + 2 * hi + 1]};
    cacc = __builtin_amdgcn_wmma_f32_16x16x4_f32(
        false, a2, false, b2, (short)0, cacc, false, false);
  }

  // D layout: VGPR r holds dens row r (lanes 0-15) / row r+8 (lanes 16-31),
  // all columns identical. Each lane-half sums log(dens) for its 8 rows.
  float s = 0.f;
#pragma unroll
  for (int r = 0; r < 8; ++r) s += __logf(cacc[r]);
  float tot = s + __shfl_xor(s, 16, 32);   // rows 0-7 + rows 8-15
  if (lane == 0) partials[blockIdx.x * 8 + wave] = tot;
}

// ---------------------------------------------------------------------------
// Kernel 3: deterministic fixed-order reduction + final scalar.
// ---------------------------------------------------------------------------
__global__ __launch_bounds__(256) void finalize(
    const float* __restrict__ partials, const float* __restrict__ a,
    float* __restrict__ out)
{
  __shared__ float red[256];
  const int t = threadIdx.x;
  float s = 0.f;
  for (int idx = t; idx < N_PART; idx += 256) s += partials[idx];
  red[t] = s;
  __syncthreads();
  for (int off = 128; off > 0; off >>= 1) {
    if (t < off) red[t] += red[t + off];
    __syncthreads();
  }
  if (t == 0) {
    float se = 0.f;
    for (int i = 0; i < N_MIX; ++i) se += expf(a[i]);
    out[0] = logf(se) - red[0] / (float)N_QRY;
  }
}

extern "C" void kernel_launch(void* const* d_in, const int* in_sizes, int n_in,
                              void* d_out, int out_size, void* d_ws, size_t ws_size,
                              hipStream_t stream) {
  (void)in_sizes; (void)n_in; (void)out_size; (void)ws_size;
  const float* u    = (const float*)d_in[0];
  const float* data = (const float*)d_in[1];
  const float* W    = (const float*)d_in[2];
  const float* b    = (const float*)d_in[3];
  const float* a    = (const float*)d_in[4];

  float* ws       = (float*)d_ws;
  float* tabs     = ws;                         // 16 * 4096 floats
  float* partials = ws + N_DIM * TAB_STRIDE;    // 8192 floats

  build_tables<<<N_DIM, 1024, 0, stream>>>(data, W, b, a, tabs);
  copnet_main<<<N_QRY / 128, 256, 0, stream>>>(u, W, b, a, tabs, partials);
  finalize<<<1, 256, 0, stream>>>(partials, a, (float*)d_out);
}